// LocalGraphEncoder_16140487099081
// MI455X (gfx1250) — compile-verified
//
#include <hip/hip_runtime.h>
#include <hip/hip_bf16.h>

#define W_  32
#define C_  1024
#define F_  128
#define E_  16384
#define ED_ 64
#define D_  256

typedef __attribute__((ext_vector_type(2))) float v2f;
typedef __attribute__((ext_vector_type(8))) float v8f;

// ---------------------------------------------------------------------------
// helpers
// ---------------------------------------------------------------------------
__device__ __forceinline__ float gelu_erf(float x) {
    return 0.5f * x * (1.0f + erff(x * 0.70710678118654752440f));
}

__device__ __forceinline__ float wave_sum(float v) {
    v += __shfl_xor(v, 16, 32);
    v += __shfl_xor(v, 8, 32);
    v += __shfl_xor(v, 4, 32);
    v += __shfl_xor(v, 2, 32);
    v += __shfl_xor(v, 1, 32);
    return v;
}

// One 256-thread block (8 wave32) computes O[16, 256] = X[16, K] @ Wg[K, 256]
// using V_WMMA_F32_16X16X4_F32. Each wave owns two 16x16 N-tiles.
// X is in LDS (row-major, pitch ldx). Wg is global row-major [K, 256].
// O is LDS, pitch ldo (raw GEMM result; bias applied by caller).
__device__ __forceinline__ void gemm_tile_f32(const float* __restrict__ Xlds, int ldx, int K,
                                              const float* __restrict__ Wg,
                                              float* __restrict__ Olds, int ldo)
{
    const int tid  = threadIdx.x;
    const int lane = tid & 31;
    const int wave = tid >> 5;
    const bool hi  = lane >= 16;
    const int l16  = lane & 15;

    #pragma unroll
    for (int t = 0; t < 2; ++t) {
        const int n0 = (wave * 2 + t) * 16;
        v8f acc = {0.f, 0.f, 0.f, 0.f, 0.f, 0.f, 0.f, 0.f};
        for (int k0 = 0; k0 < K; k0 += 4) {
            const int ka = k0 + (hi ? 2 : 0);
            // A fragment: 16x4, lanes 0-15 hold K=k0,k0+1; lanes 16-31 hold K=k0+2,k0+3
            v2f a;
            a.x = Xlds[l16 * ldx + ka];
            a.y = Xlds[l16 * ldx + ka + 1];
            // B fragment: 4x16, lane-half indexes N, VGPR indexes K within half
            v2f b;
            b.x = Wg[(size_t)ka * D_ + n0 + l16];
            b.y = Wg[(size_t)(ka + 1) * D_ + n0 + l16];
            acc = __builtin_amdgcn_wmma_f32_16x16x4_f32(
                false, a, false, b, (short)0, acc, false, false);
        }
        // C/D layout: VGPR r -> row r (lanes 0-15) / row r+8 (lanes 16-31), col = n0 + l16
        const int rbase = hi ? 8 : 0;
        const int col   = n0 + l16;
        #pragma unroll
        for (int r = 0; r < 8; ++r)
            Olds[(rbase + r) * ldo + col] = acc[r];
    }
}

// LayerNorm(+bias) then GELU over 16 rows of length 256 sitting in LDS.
// Wave w handles rows 2w, 2w+1 (8 elements per lane).
template <bool TO_GLOBAL>
__device__ __forceinline__ void ln_gelu_16rows(float* __restrict__ O, int ldo,
                                               const float* __restrict__ bias,
                                               const float* __restrict__ g,
                                               const float* __restrict__ beta,
                                               float* __restrict__ gdst)
{
    const int tid  = threadIdx.x;
    const int lane = tid & 31;
    const int wave = tid >> 5;
    #pragma unroll
    for (int rr = 0; rr < 2; ++rr) {
        const int r = wave * 2 + rr;
        float x[8];
        float s = 0.f;
        #pragma unroll
        for (int i = 0; i < 8; ++i) {
            const int c = lane + 32 * i;
            x[i] = O[r * ldo + c] + bias[c];
            s += x[i];
        }
        s = wave_sum(s);
        const float mean = s * (1.0f / 256.0f);
        float vs = 0.f;
        #pragma unroll
        for (int i = 0; i < 8; ++i) { const float d = x[i] - mean; vs += d * d; }
        vs = wave_sum(vs);
        const float rstd = rsqrtf(vs * (1.0f / 256.0f) + 1e-5f);
        #pragma unroll
        for (int i = 0; i < 8; ++i) {
            const int c = lane + 32 * i;
            const float y = (x[i] - mean) * rstd * g[c] + beta[c];
            const float z = gelu_erf(y);
            if (TO_GLOBAL) gdst[(size_t)r * D_ + c] = z;
            else           O[r * ldo + c] = z;
        }
    }
}

// ---------------------------------------------------------------------------
// Kernel 1: h = gelu(ln(node_conn @ w_node + b_node))   [W*C, 128] -> [W*C, 256]
// ---------------------------------------------------------------------------
__global__ void __launch_bounds__(256)
node_proj_kernel(const float* __restrict__ node_conn, const float* __restrict__ w_node,
                 const float* __restrict__ b_node, const float* __restrict__ g_node,
                 const float* __restrict__ beta_node, float* __restrict__ h)
{
    __shared__ float Xs[16 * F_];
    __shared__ float Os[16 * 260];
    const int tid = threadIdx.x;
    const size_t row0 = (size_t)blockIdx.x * 16;  // rows of flattened [W*C, F]

    for (int i = tid; i < 16 * F_; i += 256)
        Xs[i] = node_conn[row0 * F_ + i];
    __syncthreads();

    gemm_tile_f32(Xs, F_, F_, w_node, Os, 260);
    __syncthreads();

    ln_gelu_16rows<true>(Os, 260, b_node, g_node, beta_node, h + row0 * D_);
}

// ---------------------------------------------------------------------------
// Kernel 2: degree histogram
// ---------------------------------------------------------------------------
__global__ void deg_kernel(const long long* __restrict__ edge_index, float* __restrict__ deg)
{
    const int e = blockIdx.x * 256 + threadIdx.x;
    if (e < E_) atomicAdd(&deg[(int)edge_index[E_ + e]], 1.0f);
}

// ---------------------------------------------------------------------------
// Kernel 3 (fused): e = gelu(ln(edge_attr @ w_edge + b)); msgs = h[src]+e;
//                   agg[dst] += msgs   (e tile never leaves LDS)
// ---------------------------------------------------------------------------
__global__ void __launch_bounds__(256)
edge_fused_kernel(const float* __restrict__ edge_attr, const long long* __restrict__ edge_index,
                  const float* __restrict__ w_edge, const float* __restrict__ b_edge,
                  const float* __restrict__ g_edge, const float* __restrict__ beta_edge,
                  const float* __restrict__ h, float* __restrict__ agg)
{
    __shared__ float Xs[16 * ED_];
    __shared__ float Es[16 * 260];
    __shared__ int s_src[16];
    __shared__ int s_dst[16];

    const int tid = threadIdx.x;
    const size_t row0 = (size_t)blockIdx.x * 16;   // rows of flattened [W*E, ED]
    const int w  = (int)(row0 / E_);               // 16 rows never straddle a window
    const int e0 = (int)(row0 % E_);

    for (int i = tid; i < 16 * ED_; i += 256)
        Xs[i] = edge_attr[row0 * ED_ + i];
    if (tid < 16) {
        s_src[tid] = (int)edge_index[e0 + tid];
        s_dst[tid] = (int)edge_index[E_ + e0 + tid];
    }
    __syncthreads();

    gemm_tile_f32(Xs, ED_, ED_, w_edge, Es, 260);
    __syncthreads();

    ln_gelu_16rows<false>(Es, 260, b_edge, g_edge, beta_edge, nullptr);
    __syncthreads();

    // gather h[src] (L2-resident, 1KB rows) + scatter-add into agg[dst]
    const float* hw  = h   + (size_t)w * C_ * D_;
    float*       aw  = agg + (size_t)w * C_ * D_;
    #pragma unroll 4
    for (int r = 0; r < 16; ++r) {
        const float v = Es[r * 260 + tid] + hw[(size_t)s_src[r] * D_ + tid];
        atomicAdd(&aw[(size_t)s_dst[r] * D_ + tid], v);
    }
}

// ---------------------------------------------------------------------------
// Kernel 4: upd = gelu([h|agg/deg] @ w_up1 + b1) @ w_up2 + b2;
//           out = ln(h + upd); d_out[c] += out * mask[c] / W
// ---------------------------------------------------------------------------
__global__ void __launch_bounds__(256)
update_kernel(const float* __restrict__ h, const float* __restrict__ agg,
              const float* __restrict__ deg,
              const float* __restrict__ w_up1, const float* __restrict__ b_up1,
              const float* __restrict__ w_up2, const float* __restrict__ b_up2,
              const float* __restrict__ g_out, const float* __restrict__ beta_out,
              const float* __restrict__ channel_mask, float* __restrict__ graph)
{
    __shared__ float Xs[16 * 512];   // [h | agg/deg] concat, later cols 256..511 reused as GEMM2 out
    __shared__ float T1[16 * 260];

    const int tid = threadIdx.x;
    const size_t row0 = (size_t)blockIdx.x * 16;   // rows of flattened [W*C]

    for (int i = tid; i < 16 * D_; i += 256) {
        const int r = i >> 8, c = i & 255;
        const size_t gr = row0 + r;
        const int ch = (int)(gr % C_);
        float dg = deg[ch];
        dg = dg > 1.0f ? dg : 1.0f;
        Xs[r * 512 + c]       = h[gr * D_ + c];
        Xs[r * 512 + 256 + c] = agg[gr * D_ + c] / dg;
    }
    __syncthreads();

    gemm_tile_f32(Xs, 512, 512, w_up1, T1, 260);
    __syncthreads();

    for (int i = tid; i < 16 * D_; i += 256) {
        const int r = i >> 8, c = i & 255;
        T1[r * 260 + c] = gelu_erf(T1[r * 260 + c] + b_up1[c]);
    }
    __syncthreads();

    // GEMM2 output goes into the agg-half of Xs (free now); h-half still needed for residual.
    gemm_tile_f32(T1, 260, D_, w_up2, Xs + 256, 512);
    __syncthreads();

    // residual + LayerNorm + masked mean-over-windows (atomic accumulate)
    const int lane = tid & 31;
    const int wave = tid >> 5;
    #pragma unroll
    for (int rr = 0; rr < 2; ++rr) {
        const int r = wave * 2 + rr;
        const size_t gr = row0 + r;
        const int ch = (int)(gr % C_);
        float x[8];
        float s = 0.f;
        #pragma unroll
        for (int i = 0; i < 8; ++i) {
            const int c = lane + 32 * i;
            x[i] = Xs[r * 512 + c] + Xs[r * 512 + 256 + c] + b_up2[c];
            s += x[i];
        }
        s = wave_sum(s);
        const float mean = s * (1.0f / 256.0f);
        float vs = 0.f;
        #pragma unroll
        for (int i = 0; i < 8; ++i) { const float d = x[i] - mean; vs += d * d; }
        vs = wave_sum(vs);
        const float rstd  = rsqrtf(vs * (1.0f / 256.0f) + 1e-5f);
        const float scale = channel_mask[ch] * (1.0f / (float)W_);
        #pragma unroll
        for (int i = 0; i < 8; ++i) {
            const int c = lane + 32 * i;
            const float y = (x[i] - mean) * rstd * g_out[c] + beta_out[c];
            atomicAdd(&graph[(size_t)ch * D_ + c], y * scale);
        }
    }
}

// ---------------------------------------------------------------------------
// launch
// ---------------------------------------------------------------------------
extern "C" void kernel_launch(void* const* d_in, const int* in_sizes, int n_in,
                              void* d_out, int out_size, void* d_ws, size_t ws_size,
                              hipStream_t stream) {
    const float*     node_conn    = (const float*)d_in[0];
    const long long* edge_index   = (const long long*)d_in[1];
    const float*     edge_attr    = (const float*)d_in[2];
    const float*     channel_mask = (const float*)d_in[3];
    const float*     w_node       = (const float*)d_in[4];
    const float*     b_node       = (const float*)d_in[5];
    const float*     g_node       = (const float*)d_in[6];
    const float*     beta_node    = (const float*)d_in[7];
    const float*     w_edge       = (const float*)d_in[8];
    const float*     b_edge       = (const float*)d_in[9];
    const float*     g_edge       = (const float*)d_in[10];
    const float*     beta_edge    = (const float*)d_in[11];
    const float*     w_up1        = (const float*)d_in[12];
    const float*     b_up1        = (const float*)d_in[13];
    const float*     w_up2        = (const float*)d_in[14];
    const float*     b_up2        = (const float*)d_in[15];
    const float*     g_out        = (const float*)d_in[16];
    const float*     beta_out     = (const float*)d_in[17];

    const size_t hcd = (size_t)W_ * C_ * D_;
    float* h_buf   = (float*)d_ws;
    float* agg_buf = h_buf + hcd;
    float* deg_buf = agg_buf + hcd;

    hipMemsetAsync(agg_buf, 0, (hcd + C_) * sizeof(float), stream);   // agg + deg
    hipMemsetAsync(d_out, 0, (size_t)out_size * sizeof(float), stream);

    node_proj_kernel<<<(W_ * C_) / 16, 256, 0, stream>>>(
        node_conn, w_node, b_node, g_node, beta_node, h_buf);

    deg_kernel<<<(E_ + 255) / 256, 256, 0, stream>>>(edge_index, deg_buf);

    edge_fused_kernel<<<(W_ * E_) / 16, 256, 0, stream>>>(
        edge_attr, edge_index, w_edge, b_edge, g_edge, beta_edge, h_buf, agg_buf);

    update_kernel<<<(W_ * C_) / 16, 256, 0, stream>>>(
        h_buf, agg_buf, deg_buf, w_up1, b_up1, w_up2, b_up2,
        g_out, beta_out, channel_mask, (float*)d_out);
}